// KVG_41695542510269
// MI455X (gfx1250) — compile-verified
//
#include <hip/hip_runtime.h>
#include <stdint.h>

// ---------------------------------------------------------------------------
// Gather of top-k KV blocks (pure DMA problem, 0 FLOPs):
//   out[j] (j = (b*64+q)*8+k) = kv_flat[b, r_idx[j]]   , each block contiguous
//   block = 49*256 f32 = 50,176 bytes; 16,384 blocks total (~822 MB out).
// Roofline: ~822 MB stores + ~205..822 MB HBM loads (avg 4x reuse; the 6.4 MB
// per-batch kv slice lives in 192 MB L2)  =>  ~44-70 us floor @ 23.3 TB/s.
// Pure data movement -> Tensor Data Mover: one wave per block builds a 1-D
// tile D# and issues tensor_load_to_lds -> s_wait_tensorcnt ->
// tensor_store_from_lds. Load TH=RT (keep kv hot in L2), store TH=NT
// (write-once stream, don't evict kv).
// ---------------------------------------------------------------------------

typedef unsigned int v4u __attribute__((ext_vector_type(4)));
typedef int          v8i __attribute__((ext_vector_type(8)));
typedef int          v4i __attribute__((ext_vector_type(4)));
typedef float        v4f __attribute__((ext_vector_type(4)));

#define W2           49
#define CKV          256
#define BLK_FLOATS   (W2 * CKV)        // 12544 elements
#define BLK_BYTES    (BLK_FLOATS * 4)  // 50176 bytes (< 64KB static LDS cap)
#define VP2          128               // v * p2 (gather index range)
#define TOPK_PER_B   512               // p2 * topk = 64*8 blocks per batch

#define CPOL_RT      0                 // load: regular temporal (L2 reuse ~4x)
#define CPOL_NT      1                 // store: non-temporal streaming

// Build D# group 0 (128b): [1:0]=count=1, [63:32]=lds_addr,
// [120:64]=global_addr, [127:126]=type=2 ("image").
static __device__ __forceinline__ v4u make_g0(uint32_t lds_off, uint64_t gaddr) {
    v4u g0;
    g0.x = 1u;                                            // count=1, user mode
    g0.y = lds_off;                                       // LDS byte address
    g0.z = (uint32_t)(gaddr & 0xFFFFFFFFu);               // global_addr[31:0]
    g0.w = ((uint32_t)(gaddr >> 32) & 0x01FFFFFFu)        // global_addr[56:32]
           | 0x80000000u;                                 // type=2 at [127:126]
    return g0;
}

// D# group 1 (256b): data_size=2 (4B elems), 1-D tile:
// tensor_dim0 = tile_dim0 = 12544, tensor_dim1 = 1, tile_dim1/2 = 0 (unused),
// strides = 12544 elements (don't-care for a single row), no pad/iterate/mcast.
static __device__ __forceinline__ v8i make_g1() {
    v8i g1;
    g1[0] = 0x00020000;                                   // data_size=2 -> 4B
    g1[1] = (int)((BLK_FLOATS & 0xFFFF) << 16);           // tensor_dim0 lo16
    g1[2] = (int)((BLK_FLOATS >> 16) | (1u << 16));       // dim0 hi16 | dim1=1
    g1[3] = (int)((unsigned)BLK_FLOATS << 16);            // tile_dim0 = 12544
    g1[4] = 0;                                            // tile_dim1/2 unused
    g1[5] = BLK_FLOATS;                                   // dim0_stride lo32
    g1[6] = (int)((unsigned)BLK_FLOATS << 16);            // dim1_stride lo16
    g1[7] = 0;                                            // dim1_stride hi32
    return g1;
}

__global__ __launch_bounds__(32) void KVG_tdm_gather_kernel(
    const long long* __restrict__ r_idx,   // (n, p2, topk) flat, values in [0,128)
    const float*     __restrict__ kv,      // (n, 128, 49, 256) flat
    float*           __restrict__ out)     // (n, p2, topk, 49, 256) flat
{
    __shared__ float stage[BLK_FLOATS];    // 50,176 B staging tile

    const int j = (int)blockIdx.x;         // one 49KB block copy per workgroup
    const int b = j / TOPK_PER_B;          // batch index

    // Uniform scalar load of the gather index (SMEM path, KMcnt).
    const long long idx = r_idx[j];

    const uint64_t src = (uint64_t)kv
        + ((uint64_t)b * VP2 + (uint64_t)idx) * (uint64_t)BLK_BYTES;
    const uint64_t dst = (uint64_t)out + (uint64_t)j * (uint64_t)BLK_BYTES;

    // Low 32 bits of the flat LDS-aperture address = wave LDS byte offset.
    const uint32_t lds_off = (uint32_t)(uintptr_t)&stage[0];

    const v8i g1 = make_g1();
    const v4i gz4 = {0, 0, 0, 0};              // groups 2/3 unused (<=2D tensor)
    const v8i gz8 = {0, 0, 0, 0, 0, 0, 0, 0};  // clang-23 extra operand

    // DMA: global -> LDS (TENSORcnt++), regular-temporal load (L2 reuse).
    __builtin_amdgcn_tensor_load_to_lds(make_g0(lds_off, src), g1, gz4, gz4, gz8,
                                        CPOL_RT);
    // Tile must be resident in LDS before the store descriptor reads it.
    __builtin_amdgcn_s_wait_tensorcnt(0);
    // DMA: LDS -> global output (TENSORcnt++), non-temporal streaming store.
    __builtin_amdgcn_tensor_store_from_lds(make_g0(lds_off, dst), g1, gz4, gz4, gz8,
                                           CPOL_NT);
    __builtin_amdgcn_s_wait_tensorcnt(0);
}

// ---------------------------------------------------------------------------
// Plan-B path (compiled for histogram/fallback, not launched): coalesced b128
// loads (temporal: kv slice is 6.4MB/batch, ~4x reuse, lives in 192MB L2) and
// non-temporal b128 stores (output is write-once streaming -> don't pollute L2).
// ---------------------------------------------------------------------------
__global__ __launch_bounds__(256) void KVG_vec_gather_kernel(
    const long long* __restrict__ r_idx,
    const float*     __restrict__ kv,
    float*           __restrict__ out)
{
    const int j = (int)blockIdx.x;
    const int b = j / TOPK_PER_B;
    const long long idx = r_idx[j];

    const v4f* __restrict__ src =
        (const v4f*)(kv + ((long long)b * VP2 + idx) * BLK_FLOATS);
    v4f* __restrict__ dst = (v4f*)(out + (long long)j * BLK_FLOATS);

    #pragma unroll 4
    for (int t = (int)threadIdx.x; t < BLK_FLOATS / 4; t += 256) {
        v4f v = src[t];                          // global_load_b128 (RT)
        __builtin_nontemporal_store(v, &dst[t]); // global_store_b128 TH_NT
    }
}

extern "C" void kernel_launch(void* const* d_in, const int* in_sizes, int n_in,
                              void* d_out, int out_size, void* d_ws, size_t ws_size,
                              hipStream_t stream) {
    const long long* r_idx = (const long long*)d_in[0];  // int64 per reference
    // d_in[1] = r_weight: unused by the reference output.
    const float* kv  = (const float*)d_in[2];
    float*       out = (float*)d_out;

    const int n_blocks = in_sizes[0];  // n*p2*topk = 16384 block copies

    KVG_tdm_gather_kernel<<<dim3(n_blocks), dim3(32), 0, stream>>>(r_idx, kv, out);
}